// AttentionHead_68676527063358
// MI455X (gfx1250) — compile-verified
//
#include <hip/hip_runtime.h>
#include <hip/hip_bf16.h>
#include <math.h>

// ---------------------------------------------------------------------------
// AttentionHead (GAT-like) for MI455X / gfx1250, wave32.
//
//   item_scaled = emb @ W_scale + b_scale      (WMMA f32 16x16x4, B in LDS
//                                               via async-to-LDS staging)
//   s1[i] = item_scaled[i] . W_att[:128];  s2[i] = item_scaled[i] . W_att[128:]
//   att[e] = leaky_relu(s1[src]+s2[dst]+b_att);  score = exp(att-1)
//   score_sum[src] += score
//   agg[src] += (score/score_sum[src]) * item_scaled[dst]   (sorted segments)
//   out = sigmoid(agg)
// ---------------------------------------------------------------------------

typedef float v2f __attribute__((ext_vector_type(2)));
typedef float v8f __attribute__((ext_vector_type(8)));
typedef int   v4i __attribute__((ext_vector_type(4)));

#define GAS __attribute__((address_space(1)))
#define LAS __attribute__((address_space(3)))

#define EMB 128

#define HAVE_ASYNC_LDS 0
#if defined(__has_builtin)
#if __has_builtin(__builtin_amdgcn_global_load_async_to_lds_b128)
#undef HAVE_ASYNC_LDS
#define HAVE_ASYNC_LDS 1
#endif
#endif

#define HAVE_WAIT_ASYNC 0
#if defined(__has_builtin)
#if __has_builtin(__builtin_amdgcn_s_wait_asynccnt)
#undef HAVE_WAIT_ASYNC
#define HAVE_WAIT_ASYNC 1
#endif
#endif

// ---------------- WMMA GEMM: C[M x 128] = A[M x 128] * B[128 x 128] + bias --
// Block = 256 threads (8 waves) handles a 64-row strip. W_scale (64 KB) is
// staged into LDS once per block with async-to-LDS b128 transfers. Each wave
// owns one 16-col N-tile and computes four 16x16 M-tiles, reusing each B
// fragment 4x. K-loop: 32 steps of V_WMMA_F32_16X16X4_F32 per tile.
__global__ __launch_bounds__(256) void gemm_wmma_f32(
    const float* __restrict__ A, const float* __restrict__ B,
    const float* __restrict__ bias, float* __restrict__ C, int M) {
  __shared__ float lds_B[EMB * EMB];           // 64 KB

  // ---- stage B into LDS (4096 float4 across 256 threads) ----
  for (int i = threadIdx.x; i < EMB * EMB / 4; i += 256) {
#if HAVE_ASYNC_LDS
    __builtin_amdgcn_global_load_async_to_lds_b128(
        (GAS v4i*)((const float4*)B + i),
        (LAS v4i*)((float4*)lds_B + i),
        0, 0);
#else
    ((float4*)lds_B)[i] = ((const float4*)B)[i];
#endif
  }
#if HAVE_ASYNC_LDS
#if HAVE_WAIT_ASYNC
  __builtin_amdgcn_s_wait_asynccnt(0);
#else
  asm volatile("s_wait_asynccnt 0" ::: "memory");
#endif
#endif
  __syncthreads();

  const int wave = threadIdx.x >> 5;           // 0..7 -> N tile
  const int lane = threadIdx.x & 31;
  const int half = lane >> 4;                  // 0: lanes 0-15, 1: lanes 16-31
  const int l16  = lane & 15;

  const int row0 = blockIdx.x * 64;
  const int col0 = wave * 16;

  // Clamped A row pointers (no predication: garbage rows never stored)
  const float* Arow[4];
#pragma unroll
  for (int mt = 0; mt < 4; ++mt) {
    int r = row0 + 16 * mt + l16;
    r = (r < M) ? r : (M - 1);
    Arow[mt] = A + (size_t)r * EMB;
  }

  v8f acc[4] = {};
  for (int k0 = 0; k0 < EMB; k0 += 4) {
    // B 4x16 fragment from LDS: VGPR0: K=k0 (lo)/k0+2 (hi); VGPR1: +1
    v2f b;
    b.x = lds_B[(k0 + 2 * half)     * EMB + col0 + l16];
    b.y = lds_B[(k0 + 2 * half + 1) * EMB + col0 + l16];
#pragma unroll
    for (int mt = 0; mt < 4; ++mt) {
      // A 16x4 fragment: contiguous pair -> single b64 load
      const v2f a = *(const v2f*)(Arow[mt] + k0 + 2 * half);
      acc[mt] = __builtin_amdgcn_wmma_f32_16x16x4_f32(
          false, a, false, b, (short)0, acc[mt], false, false);
    }
  }

  // Epilogue: VGPR g -> row = row0 + 16*mt + g + 8*half, col = col0 + l16.
  // Scalar (wave-uniform) branch: all blocks except the last store a full
  // 64-row strip with no per-store predication.
  const float bv = bias[col0 + l16];
  float* Cbase = C + (size_t)row0 * EMB + col0 + l16;
  if (row0 + 64 <= M) {
#pragma unroll
    for (int mt = 0; mt < 4; ++mt) {
#pragma unroll
      for (int g = 0; g < 8; ++g) {
        Cbase[(size_t)(16 * mt + g + 8 * half) * EMB] = acc[mt][g] + bv;
      }
    }
  } else {
#pragma unroll
    for (int mt = 0; mt < 4; ++mt) {
#pragma unroll
      for (int g = 0; g < 8; ++g) {
        const int r = row0 + 16 * mt + g + 8 * half;
        if (r < M) C[(size_t)r * EMB + col0 + l16] = acc[mt][g] + bv;
      }
    }
  }
}

// ---------------- per-item attention scalars: s1, s2 -----------------------
__global__ __launch_bounds__(256) void row_dots_kernel(
    const float* __restrict__ IS, const float* __restrict__ W_att,
    float* __restrict__ s1, float* __restrict__ s2, int M) {
  const int wave = (blockIdx.x * blockDim.x + threadIdx.x) >> 5;  // row
  const int lane = threadIdx.x & 31;
  if (wave >= M) return;
  const float* row = IS + (size_t)wave * EMB;
  float d1 = 0.0f, d2 = 0.0f;
#pragma unroll
  for (int j = 0; j < EMB; j += 32) {
    const float v = row[j + lane];
    d1 += v * W_att[j + lane];
    d2 += v * W_att[EMB + j + lane];
  }
#pragma unroll
  for (int off = 16; off > 0; off >>= 1) {
    d1 += __shfl_down(d1, off, 32);
    d2 += __shfl_down(d2, off, 32);
  }
  if (lane == 0) { s1[wave] = d1; s2[wave] = d2; }
}

// ---------------- zero init -------------------------------------------------
__global__ void zero_kernel(float* __restrict__ p, int n) {
  const int i = blockIdx.x * blockDim.x + threadIdx.x;
  if (i < n) p[i] = 0.0f;
}

// ---------------- per-edge scores + segment sums ----------------------------
__global__ __launch_bounds__(256) void edge_score_kernel(
    const int* __restrict__ edge, const float* __restrict__ s1,
    const float* __restrict__ s2, const float* __restrict__ b_att,
    float* __restrict__ score, float* __restrict__ ssum, int E) {
  const int e = blockIdx.x * blockDim.x + threadIdx.x;
  if (e >= E) return;
  const int src = edge[2 * e];
  const int dst = edge[2 * e + 1];
  float att = s1[src] + s2[dst] + b_att[0];
  att = (att > 0.0f) ? att : 0.2f * att;          // leaky_relu(0.2)
  const float sc = expf(att - 1.0f);
  score[e] = sc;
  atomicAdd(&ssum[src], sc);
}

// ---------------- weighted aggregation (edges sorted by src) ----------------
// One wave handles EPW consecutive edges; each lane owns 4 columns (float4).
// Accumulate in registers while src is unchanged; flush with 4 atomics per
// lane only at segment boundaries (avg segment length ~32 edges).
#define EPW 64
__global__ __launch_bounds__(256) void aggregate_kernel(
    const int* __restrict__ edge, const float* __restrict__ score,
    const float* __restrict__ ssum, const float* __restrict__ IS,
    float* __restrict__ agg, int E) {
  const int wave = (blockIdx.x * blockDim.x + threadIdx.x) >> 5;
  const int lane = threadIdx.x & 31;
  const int e0 = wave * EPW;
  if (e0 >= E) return;
  const int e1 = (e0 + EPW < E) ? (e0 + EPW) : E;

  float4 acc = make_float4(0.f, 0.f, 0.f, 0.f);
  int cur = edge[2 * e0];
  float inv = 1.0f / ssum[cur];

  for (int e = e0; e < e1; ++e) {
    const int src = edge[2 * e];
    const int dst = edge[2 * e + 1];
    // prefetch a future edge's item row into cache (global_prefetch_b8)
    if (e + 8 < E) {
      __builtin_prefetch(IS + (size_t)edge[2 * (e + 8) + 1] * EMB, 0, 1);
    }
    if (src != cur) {
      float* dst_p = agg + (size_t)cur * EMB + lane * 4;
      atomicAdd(dst_p + 0, acc.x);
      atomicAdd(dst_p + 1, acc.y);
      atomicAdd(dst_p + 2, acc.z);
      atomicAdd(dst_p + 3, acc.w);
      acc = make_float4(0.f, 0.f, 0.f, 0.f);
      cur = src;
      inv = 1.0f / ssum[cur];
    }
    const float w = score[e] * inv;
    const float4 it = *(const float4*)(IS + (size_t)dst * EMB + lane * 4);
    acc.x += w * it.x;
    acc.y += w * it.y;
    acc.z += w * it.z;
    acc.w += w * it.w;
  }
  float* dst_p = agg + (size_t)cur * EMB + lane * 4;
  atomicAdd(dst_p + 0, acc.x);
  atomicAdd(dst_p + 1, acc.y);
  atomicAdd(dst_p + 2, acc.z);
  atomicAdd(dst_p + 3, acc.w);
}

// ---------------- sigmoid epilogue (in place on d_out) ----------------------
__global__ void sigmoid_kernel(float* __restrict__ p, int n) {
  const int i = blockIdx.x * blockDim.x + threadIdx.x;
  if (i < n) {
    const float x = p[i];
    p[i] = 1.0f / (1.0f + expf(-x));
  }
}

// ---------------------------------------------------------------------------
extern "C" void kernel_launch(void* const* d_in, const int* in_sizes, int n_in,
                              void* d_out, int out_size, void* d_ws, size_t ws_size,
                              hipStream_t stream) {
  const float* emb   = (const float*)d_in[0];   // (M, 128)
  const int*   edge  = (const int*)  d_in[1];   // (E, 2)
  const float* Wsc   = (const float*)d_in[2];   // (128, 128)
  const float* bsc   = (const float*)d_in[3];   // (128,)
  const float* Watt  = (const float*)d_in[4];   // (256, 1)
  const float* batt  = (const float*)d_in[5];   // (1,)
  float* out = (float*)d_out;                   // (M, 128)

  const int M = in_sizes[0] / EMB;              // 20001
  const int E = in_sizes[1] / 2;                // 640000

  // workspace layout (floats)
  float* ws = (float*)d_ws;
  float* item_scaled = ws;                         // M*128
  float* s1    = item_scaled + (size_t)M * EMB;    // M
  float* s2    = s1 + M;                           // M
  float* score = s2 + M;                           // E
  float* ssum  = score + E;                        // M

  // zero accumulators (out is used for agg)
  zero_kernel<<<(M * EMB + 255) / 256, 256, 0, stream>>>(out, M * EMB);
  zero_kernel<<<(M + 255) / 256, 256, 0, stream>>>(ssum, M);

  // 1) item_scaled = emb @ W_scale + b_scale   (WMMA, B staged in LDS)
  gemm_wmma_f32<<<(M + 63) / 64, 256, 0, stream>>>(emb, Wsc, bsc, item_scaled, M);

  // 2) s1/s2 per item (one wave per row)
  row_dots_kernel<<<(M * 32 + 255) / 256, 256, 0, stream>>>(item_scaled, Watt, s1, s2, M);

  // 3) edge scores + segment sums
  edge_score_kernel<<<(E + 255) / 256, 256, 0, stream>>>(edge, s1, s2, batt, score, ssum, E);

  // 4) weighted aggregation into out
  {
    const int waves = (E + EPW - 1) / EPW;
    const int blocks = (waves * 32 + 255) / 256;
    aggregate_kernel<<<blocks, 256, 0, stream>>>(edge, score, ssum, item_scaled, out, E);
  }

  // 5) sigmoid in place
  sigmoid_kernel<<<(M * EMB + 255) / 256, 256, 0, stream>>>(out, M * EMB);
}